// PivotMapPostProcessor_6923487281680
// MI455X (gfx1250) — compile-verified
//
#include <hip/hip_runtime.h>
#include <hip/hip_bf16.h>
#include <math.h>

typedef float v2f __attribute__((ext_vector_type(2)));
typedef float v8f __attribute__((ext_vector_type(8)));

#define NQ    100        // detections D
#define NG    30         // ground truths G
#define NPIX  80000      // pixels P = 400*200
#define NPTS  20         // m == n == 20
#define BIGF  1e9f
#define DTILES 7         // ceil(100/16)
#define KPARTS 40
#define QPP   (NPIX / 4 / KPARTS)   // 500 K-quads per partition

// ---- workspace layout (floats) ----
#define WS_XY    0            // [NQ][NG]  x @ y^T            (atomic-accumulated)
#define WS_PY    3000         // [NQ][NG]  sigmoid(x) @ y^T   (atomic-accumulated)
#define WS_CONST 6000         // [NQ] sum relu(x)+softplus(-|x|)
#define WS_PSUM  6112         // [NQ] sum sigmoid(x)
#define WS_YSUM  6224         // [NG] sum y
#define WS_SEQ   6272         // [NQ][NG] seq-match cost
#define WS_ZERO_N 6000        // only the atomic accumulators need zeroing

__device__ __forceinline__ float fast_rcp(float x) {
    return __builtin_amdgcn_rcpf(x);        // single v_rcp_f32
}

__global__ __launch_bounds__(256)
void zero_ws_kernel(float* ws, int n) {
    int i = blockIdx.x * blockDim.x + threadIdx.x;
    if (i < n) ws[i] = 0.0f;
}

// blocks 0..NQ-1: per-detection reductions over x; blocks NQ..NQ+NG-1: y row sums
__global__ __launch_bounds__(256)
void row_reduce_kernel(const float* __restrict__ x, const float* __restrict__ y,
                       float* __restrict__ ws) {
    __shared__ float s1[256];
    __shared__ float s2[256];
    const int b = blockIdx.x;
    const int t = threadIdx.x;
    float a1 = 0.0f, a2 = 0.0f;
    if (b < NQ) {
        const float* r = x + (long long)b * NPIX;
        for (int i = t; i < NPIX; i += 256) {
            float v  = r[i];
            float av = fabsf(v);
            a1 += fmaxf(v, 0.0f) + __logf(1.0f + __expf(-av)); // relu + softplus(-|v|)
            a2 += fast_rcp(1.0f + __expf(-v));                 // sigmoid
        }
    } else {
        const float* r = y + (long long)(b - NQ) * NPIX;
        for (int i = t; i < NPIX; i += 256) a1 += r[i];
    }
    s1[t] = a1; s2[t] = a2;
    __syncthreads();
    for (int s = 128; s > 0; s >>= 1) {
        if (t < s) { s1[t] += s1[t + s]; s2[t] += s2[t + s]; }
        __syncthreads();
    }
    if (t == 0) {
        if (b < NQ) { ws[WS_CONST + b] = s1[0]; ws[WS_PSUM + b] = s2[0]; }
        else        { ws[WS_YSUM + (b - NQ)] = s1[0]; }
    }
}

// grid: (DTILES, KPARTS); block: 64 threads = 2 waves, wave w handles G-tile w.
// Computes partial xy = x@y^T and py = sigmoid(x)@y^T via V_WMMA_F32_16X16X4_F32,
// atomically accumulated into workspace.
__global__ __launch_bounds__(64)
void gemm_wmma_kernel(const float* __restrict__ x, const float* __restrict__ y,
                      float* __restrict__ ws) {
    const int lane  = threadIdx.x & 31;
    const int gtile = threadIdx.x >> 5;      // 0 or 1
    const int dtile = blockIdx.x;
    const int kpart = blockIdx.y;
    const int hgrp  = lane >> 4;             // lane half: selects K pair {0,1} vs {2,3}
    const int l15   = lane & 15;

    const int row = dtile * 16 + l15;        // A row this lane feeds (M = lane%16)
    const int col = gtile * 16 + l15;        // B col this lane feeds (N = lane%16)
    const float rs = (row < NQ) ? 1.0f : 0.0f;
    const float cs = (col < NG) ? 1.0f : 0.0f;
    const float* xr = x + (long long)((row < NQ) ? row : 0) * NPIX;
    const float* yr = y + (long long)((col < NG) ? col : 0) * NPIX;

    v8f acc_xy = {};
    v8f acc_py = {};
    const int q0 = kpart * QPP;
    const int q1 = q0 + QPP;
#pragma unroll 4
    for (int q = q0; q < q1; ++q) {
        const int k = q * 4 + hgrp * 2;      // lanes 0-15 -> K={0,1}; 16-31 -> K={2,3}
        v2f a = *(const v2f*)(xr + k);
        v2f b = *(const v2f*)(yr + k);
        a *= rs;
        b *= cs;
        v2f sa;
        sa.x = rs * fast_rcp(1.0f + __expf(-a.x));   // masked sigmoid, v_rcp_f32 path
        sa.y = rs * fast_rcp(1.0f + __expf(-a.y));
        acc_xy = __builtin_amdgcn_wmma_f32_16x16x4_f32(
            false, a,  false, b, (short)0, acc_xy, false, false);
        acc_py = __builtin_amdgcn_wmma_f32_16x16x4_f32(
            false, sa, false, b, (short)0, acc_py, false, false);
    }

    // C/D layout: VGPR r, lanes 0-15 -> M=r, N=lane; lanes 16-31 -> M=8+r, N=lane-16
    float* xy = ws + WS_XY;
    float* py = ws + WS_PY;
#pragma unroll
    for (int r = 0; r < 8; ++r) {
        const int orow = dtile * 16 + r + 8 * hgrp;
        const int ocol = gtile * 16 + l15;
        if (orow < NQ && ocol < NG) {
            atomicAdd(&xy[orow * NG + ocol], acc_xy[r]);
            atomicAdd(&py[orow * NG + ocol], acc_py[r]);
        }
    }
}

// One thread per (d, g): monotone sequence-matching DP, m=n=20.
__global__ __launch_bounds__(128)
void seq_dp_kernel(const float* __restrict__ dtp, const float* __restrict__ gtp,
                   const int* __restrict__ valid_len, float* __restrict__ ws) {
    const int idx = blockIdx.x * blockDim.x + threadIdx.x;
    if (idx >= NQ * NG) return;
    const int d = idx / NG;
    const int g = idx % NG;
    const int len = valid_len[g];

    float dx[NPTS], dy[NPTS];
    const float* dr = dtp + d * (NPTS * 2);
#pragma unroll
    for (int j = 0; j < NPTS; ++j) { dx[j] = dr[2 * j]; dy[j] = dr[2 * j + 1]; }
    const float* gr = gtp + g * (NPTS * 2);

    float dp[NPTS];
    {   // row 0: dp[0] = w0 * c(0,0); rest BIG
        const float gx = gr[0], gy = gr[1];
        const float w0 = 2.0f + ((0 == len - 1) ? 1.0f : 0.0f);
        dp[0] = w0 * (fabsf(gx - dx[0]) + fabsf(gy - dy[0]));
#pragma unroll
        for (int j = 1; j < NPTS; ++j) dp[j] = BIGF;
    }
    for (int i = 1; i < NPTS; ++i) {
        if (i < len) {
            const float gx = gr[2 * i], gy = gr[2 * i + 1];
            const float w = 1.0f + ((i == len - 1) ? 1.0f : 0.0f);
            float run = BIGF;              // min(dp[0..j-1]); BIG for j==0
            float nd[NPTS];
#pragma unroll
            for (int j = 0; j < NPTS; ++j) {
                const float c = w * (fabsf(gx - dx[j]) + fabsf(gy - dy[j]));
                nd[j] = c + run;
                run = fminf(run, dp[j]);
            }
#pragma unroll
            for (int j = 0; j < NPTS; ++j) dp[j] = nd[j];
        }
    }
    ws[WS_SEQ + idx] = dp[NPTS - 1] / ((float)len + 2.0f);
}

__global__ __launch_bounds__(128)
void combine_kernel(const float* __restrict__ obj, const int* __restrict__ labels,
                    const float* __restrict__ ws, float* __restrict__ out) {
    const int idx = blockIdx.x * blockDim.x + threadIdx.x;
    if (idx >= NQ * NG) return;
    const int d = idx / NG;
    const int g = idx % NG;

    // classification: -softmax(obj[d])[label[g]]  (exact IEEE division here)
    const float l0 = obj[2 * d], l1 = obj[2 * d + 1];
    const float mx = fmaxf(l0, l1);
    const float e0 = __expf(l0 - mx), e1 = __expf(l1 - mx);
    const float prob = ((labels[g] == 0) ? e0 : e1) / (e0 + e1);
    const float cost_obj = -prob;

    // mask: BCE + dice
    const float bce   = (ws[WS_CONST + d] - ws[WS_XY + idx]) / (float)NPIX;
    const float inter = ws[WS_PY + idx];
    const float dice  = 1.0f - (2.0f * inter + 1.0f) /
                               (ws[WS_PSUM + d] + ws[WS_YSUM + g] + 1.0f);
    const float cost_mask = bce + dice;

    out[idx] = cost_obj + cost_mask + 2.0f * ws[WS_SEQ + idx];
}

extern "C" void kernel_launch(void* const* d_in, const int* in_sizes, int n_in,
                              void* d_out, int out_size, void* d_ws, size_t ws_size,
                              hipStream_t stream) {
    const float* obj_logits = (const float*)d_in[0];   // [1,100,2]
    const float* mask_log   = (const float*)d_in[1];   // [1,100,400,200]
    const float* ctr_pts    = (const float*)d_in[2];   // [1,100,20,2]
    const float* gt_pts     = (const float*)d_in[3];   // [30,20,2]
    const float* gt_masks   = (const float*)d_in[4];   // [30,400,200]
    const int*   gt_labels  = (const int*)d_in[5];     // [30]
    const int*   valid_len  = (const int*)d_in[6];     // [30]
    float* out = (float*)d_out;                        // [1,100,30]
    float* ws  = (float*)d_ws;

    zero_ws_kernel<<<(WS_ZERO_N + 255) / 256, 256, 0, stream>>>(ws, WS_ZERO_N);
    row_reduce_kernel<<<NQ + NG, 256, 0, stream>>>(mask_log, gt_masks, ws);
    gemm_wmma_kernel<<<dim3(DTILES, KPARTS), 64, 0, stream>>>(mask_log, gt_masks, ws);
    seq_dp_kernel<<<(NQ * NG + 127) / 128, 128, 0, stream>>>(ctr_pts, gt_pts, valid_len, ws);
    combine_kernel<<<(NQ * NG + 127) / 128, 128, 0, stream>>>(obj_logits, gt_labels, ws, out);
}